// Model_2997887173067
// MI455X (gfx1250) — compile-verified
//
#include <hip/hip_runtime.h>
#include <math.h>

typedef __attribute__((ext_vector_type(2))) float v2f;
typedef __attribute__((ext_vector_type(8))) float v8f;

#define TT   1024
#define NN   512
#define BB   16
#define IND  1536
#define NS_ITERS 24

__device__ inline v8f wmma4(v2f a, v2f b, v8f c) {
  // V_WMMA_F32_16X16X4_F32 : D = A(16x4) * B(4x16) + C(16x16)
  return __builtin_amdgcn_wmma_f32_16x16x4_f32(
      false, a, false, b, (short)0, c, false, false);
}

__device__ inline float block_reduce_sum(float v) {
  __shared__ float sm[256];
  int t = threadIdx.x;
  sm[t] = v; __syncthreads();
  for (int s = 128; s > 0; s >>= 1) { if (t < s) sm[t] += sm[t + s]; __syncthreads(); }
  float r = sm[0]; __syncthreads();
  return r;
}

// ---------------- WMMA GEMM: C[b] = alpha*op(A[b])@B[b] (+beta*E) (+bias) (relu) ----
// op(A)=A (MxK, lda) if !AT ; if AT, A stored KxM with leading dim lda.
#define BM 128
#define BN 64
#define BK 16

template<bool AT, bool HAS_E, bool HAS_BIAS, bool RELU>
__global__ __launch_bounds__(256) void gemm_wmma(
    const float* __restrict__ Ag, int lda, long long sA,
    const float* __restrict__ Bg, int ldb, long long sB,
    float* __restrict__ Cg, int ldc, long long sC,
    const float* __restrict__ Eg, long long sE,
    const float* __restrict__ bias,
    float alpha, float beta, int K)
{
  // Both tiles stored K-minor with 17-float stride: frag = contiguous dword
  // pair (single ds_load_2addr), and 17*lane mod 64 banks are all distinct.
  __shared__ float As[BM][BK + 1];   // [m][k]
  __shared__ float Bs[BN][BK + 1];   // [n][k] (transposed on store)

  const int b = blockIdx.z;
  const int mBase = blockIdx.y * BM;
  const int nBase = blockIdx.x * BN;
  const float* A = Ag + (long long)b * sA;
  const float* B = Bg + (long long)b * sB;
  float* C = Cg + (long long)b * sC;

  const int tid  = threadIdx.x;
  const int lane = tid & 31;
  const int wave = tid >> 5;
  const int nl = lane & 15;
  const int kh = lane >> 4;            // 0/1 : K half per ISA f32 layout
  const int wm = (wave >> 1) * 32;
  const int wn = (wave & 1) * 32;

  v8f c00 = {}; v8f c01 = {}; v8f c10 = {}; v8f c11 = {};

  for (int k0 = 0; k0 < K; k0 += BK) {
    // ---- stage A tile (BM x BK) ----
    if (AT) {
      int mm = (tid & 31) * 4;
      int kk = tid >> 5;               // 0..7
#pragma unroll
      for (int r = 0; r < 2; ++r) {
        int k = kk + r * 8;
        const float4 v = *(const float4*)(A + (long long)(k0 + k) * lda + mBase + mm);
        As[mm + 0][k] = v.x; As[mm + 1][k] = v.y; As[mm + 2][k] = v.z; As[mm + 3][k] = v.w;
      }
    } else {
      int i  = tid >> 1;               // 0..127
      int jj = (tid & 1) * 8;
      const float* src = A + (long long)(mBase + i) * lda + k0 + jj;
      float4 v0 = *(const float4*)(src);
      float4 v1 = *(const float4*)(src + 4);
      As[i][jj+0]=v0.x; As[i][jj+1]=v0.y; As[i][jj+2]=v0.z; As[i][jj+3]=v0.w;
      As[i][jj+4]=v1.x; As[i][jj+5]=v1.y; As[i][jj+6]=v1.z; As[i][jj+7]=v1.w;
    }
    // ---- stage B tile (BK x BN), transposed into Bs[n][k] ----
    {
      int kk = tid >> 4;               // 0..15
      int nn = (tid & 15) * 4;
      const float4 v = *(const float4*)(B + (long long)(k0 + kk) * ldb + nBase + nn);
      Bs[nn + 0][kk] = v.x; Bs[nn + 1][kk] = v.y; Bs[nn + 2][kk] = v.z; Bs[nn + 3][kk] = v.w;
    }
    __syncthreads();

    // ---- prefetch ALL fragments for this BK chunk, then 16 WMMAs back-to-back ----
    v2f af[2][4], bf[2][4];
#pragma unroll
    for (int s4 = 0; s4 < 4; ++s4) {
      int k = s4 * 4 + 2 * kh;
      af[0][s4].x = As[wm + nl][k];        af[0][s4].y = As[wm + nl][k + 1];
      af[1][s4].x = As[wm + 16 + nl][k];   af[1][s4].y = As[wm + 16 + nl][k + 1];
      bf[0][s4].x = Bs[wn + nl][k];        bf[0][s4].y = Bs[wn + nl][k + 1];
      bf[1][s4].x = Bs[wn + 16 + nl][k];   bf[1][s4].y = Bs[wn + 16 + nl][k + 1];
    }
#pragma unroll
    for (int s4 = 0; s4 < 4; ++s4) {
      c00 = wmma4(af[0][s4], bf[0][s4], c00);
      c01 = wmma4(af[0][s4], bf[1][s4], c01);
      c10 = wmma4(af[1][s4], bf[0][s4], c10);
      c11 = wmma4(af[1][s4], bf[1][s4], c11);
    }
    __syncthreads();
  }

  const float* E = HAS_E ? (Eg + (long long)b * sE) : nullptr;
#pragma unroll
  for (int r = 0; r < 8; ++r) {
    int rows[2] = { wm + r + 8 * kh, wm + r + 8 * kh + 16 };
    int cols[2] = { wn + nl, wn + nl + 16 };
    float vals[2][2] = { { c00[r], c01[r] }, { c10[r], c11[r] } };
#pragma unroll
    for (int rr = 0; rr < 2; ++rr)
#pragma unroll
      for (int cc = 0; cc < 2; ++cc) {
        long long gi = (long long)(mBase + rows[rr]) * ldc + (nBase + cols[cc]);
        float v = alpha * vals[rr][cc];
        if constexpr (HAS_E)    v += beta * E[gi];
        if constexpr (HAS_BIAS) v += bias[nBase + cols[cc]];
        if constexpr (RELU)     v = fmaxf(v, 0.f);
        C[gi] = v;
      }
  }
}

// ---------------- stage kernels ----------------
__global__ __launch_bounds__(256) void mean_kernel(const float* __restrict__ x, float* mu) {
  __shared__ float red[4][64];
  int nl = threadIdx.x & 63, tg = threadIdx.x >> 6;
  int n = blockIdx.x * 64 + nl, b = blockIdx.y;
  const float* xp = x + ((long long)b * TT) * NN + n;
  float s = 0.f;
  for (int t = tg; t < TT; t += 4) s += xp[(long long)t * NN];
  red[tg][nl] = s; __syncthreads();
  if (tg == 0) mu[b * NN + n] = (red[0][nl] + red[1][nl] + red[2][nl] + red[3][nl]) / (float)TT;
}

__global__ __launch_bounds__(256) void center_kernel(const float* __restrict__ x,
                                                     const float* __restrict__ mu,
                                                     float* __restrict__ xm) {
  long long total = (long long)BB * TT * NN;
  for (long long idx = blockIdx.x * 256ll + threadIdx.x; idx < total;
       idx += (long long)gridDim.x * 256) {
    int n = (int)(idx % NN);
    int b = (int)(idx / ((long long)TT * NN));
    xm[idx] = x[idx] - mu[b * NN + n];
  }
}

// counting rank == double-argsort (stable) ; write centered ranks
__global__ __launch_bounds__(256) void rank_kernel(const float* __restrict__ x,
                                                   float* __restrict__ rm) {
  __shared__ float s[TT];
  int b = blockIdx.y, n = blockIdx.x;
  const float* xp = x + ((long long)b * TT) * NN + n;
  for (int t = threadIdx.x; t < TT; t += 256) s[t] = xp[(long long)t * NN];
  __syncthreads();
  for (int i = threadIdx.x; i < TT; i += 256) {
    float xi = s[i]; int cnt = 0;
    for (int j = 0; j < TT; ++j) {
      float xj = s[j];
      cnt += (xj < xi) || (xj == xi && j < i);
    }
    rm[((long long)b * TT + i) * NN + n] = (float)(cnt + 1) - 0.5f * (TT + 1);
  }
}

__global__ __launch_bounds__(256) void diag_sd_kernel(const float* __restrict__ M, float* sd) {
  int n = blockIdx.x * 256 + threadIdx.x, b = blockIdx.y;
  sd[b * NN + n] = sqrtf(fmaxf(M[((long long)b * NN + n) * NN + n], 1e-8f));
}

__global__ __launch_bounds__(256) void corr_norm_kernel(float* __restrict__ M,
                                                        const float* __restrict__ sd) {
  long long total = (long long)BB * NN * NN;
  for (long long idx = blockIdx.x * 256ll + threadIdx.x; idx < total;
       idx += (long long)gridDim.x * 256) {
    int j = (int)(idx % NN);
    int i = (int)((idx / NN) % NN);
    int b = (int)(idx / ((long long)NN * NN));
    float v = M[idx] / (sd[b * NN + i] * sd[b * NN + j]);
    if (!isfinite(v)) v = 0.f;
    M[idx] = (i == j) ? 1.f : v;
  }
}

__global__ __launch_bounds__(256) void ridge_kernel(const float* __restrict__ cov,
                                                    float* __restrict__ A) {
  long long total = (long long)BB * NN * NN;
  for (long long idx = blockIdx.x * 256ll + threadIdx.x; idx < total;
       idx += (long long)gridDim.x * 256) {
    int j = (int)(idx % NN);
    int i = (int)((idx / NN) % NN);
    A[idx] = cov[idx] + ((i == j) ? 1e-3f : 0.f);
  }
}

__global__ __launch_bounds__(256) void rowsum_abs_kernel(const float* __restrict__ A, float* rs) {
  int b = blockIdx.y, i = blockIdx.x;
  const float* row = A + ((long long)b * NN + i) * NN;
  float s = 0.f;
  for (int j = threadIdx.x; j < NN; j += 256) s += fabsf(row[j]);
  float r = block_reduce_sum(s);
  if (threadIdx.x == 0) rs[b * NN + i] = r;
}

__global__ __launch_bounds__(256) void batch_scale_kernel(const float* __restrict__ rs, float* scl) {
  __shared__ float sm[256];
  int b = blockIdx.x;
  float m = 0.f;
  for (int i = threadIdx.x; i < NN; i += 256) m = fmaxf(m, rs[b * NN + i]);
  sm[threadIdx.x] = m; __syncthreads();
  for (int s = 128; s > 0; s >>= 1) { if (threadIdx.x < s) sm[threadIdx.x] = fmaxf(sm[threadIdx.x], sm[threadIdx.x + s]); __syncthreads(); }
  if (threadIdx.x == 0) scl[b] = 1.f / (sm[0] * sm[0]);
}

__global__ __launch_bounds__(256) void scale_init_kernel(const float* __restrict__ A,
                                                         const float* __restrict__ scl,
                                                         float* __restrict__ X0) {
  long long total = (long long)BB * NN * NN;
  for (long long idx = blockIdx.x * 256ll + threadIdx.x; idx < total;
       idx += (long long)gridDim.x * 256) {
    int b = (int)(idx / ((long long)NN * NN));
    X0[idx] = A[idx] * scl[b];
  }
}

__global__ __launch_bounds__(256) void pcorr_kernel(const float* __restrict__ prec,
                                                    const float* __restrict__ sd,
                                                    float* __restrict__ Cm) {
  long long total = (long long)BB * NN * NN;
  for (long long idx = blockIdx.x * 256ll + threadIdx.x; idx < total;
       idx += (long long)gridDim.x * 256) {
    int j = (int)(idx % NN);
    int i = (int)((idx / NN) % NN);
    int b = (int)(idx / ((long long)NN * NN));
    float v = (i == j) ? 1.f : (-prec[idx] / (sd[b * NN + i] * sd[b * NN + j]));
    Cm[idx] = tanhf(v);
  }
}

__global__ void zero_kernel(float* p, int n) {
  int i = blockIdx.x * 256 + threadIdx.x;
  if (i < n) p[i] = 0.f;
}

__global__ __launch_bounds__(256) void fusion_stats_kernel(const float* __restrict__ P,
                                                           const float* __restrict__ S,
                                                           const float* __restrict__ Cm,
                                                           float* __restrict__ stats) {
  int b = blockIdx.y;
  long long base = (long long)b * NN * NN;
  float acc[9] = {0,0,0,0,0,0,0,0,0};
  int total = NN * NN;
  for (int idx = blockIdx.x * 256 + threadIdx.x; idx < total; idx += gridDim.x * 256) {
    int i = idx / NN, j = idx % NN;
    if (i == j) continue;
    float p = P[base + idx], s = S[base + idx], c = Cm[base + idx];
    float t = p + s + c;
    float op = 0.5f * (t - p), os = 0.5f * (t - s), oc = 0.5f * (t - c);
    acc[0] += p * op; acc[1] += p * p; acc[2] += op * op;
    acc[3] += s * os; acc[4] += s * s; acc[5] += os * os;
    acc[6] += c * oc; acc[7] += c * c; acc[8] += oc * oc;
  }
#pragma unroll
  for (int q = 0; q < 9; ++q) {
    float r = block_reduce_sum(acc[q]);
    if (threadIdx.x == 0) atomicAdd(&stats[b * 9 + q], r);
  }
}

__global__ void alpha_kernel(const float* __restrict__ stats,
                             const float* __restrict__ gamma,
                             float* __restrict__ alph) {
  int b = threadIdx.x;
  if (b >= BB) return;
  float g = gamma[0];
  float sc[3];
  for (int v = 0; v < 3; ++v) {
    float num = stats[b * 9 + v * 3 + 0];
    float nv  = fmaxf(sqrtf(stats[b * 9 + v * 3 + 1]), 1e-8f);
    float no  = fmaxf(sqrtf(stats[b * 9 + v * 3 + 2]), 1e-8f);
    sc[v] = g * (num / (nv * no));
  }
  float m = fmaxf(sc[0], fmaxf(sc[1], sc[2]));
  float e0 = expf(sc[0] - m), e1 = expf(sc[1] - m), e2 = expf(sc[2] - m);
  float inv = 1.f / (e0 + e1 + e2);
  alph[b * 3 + 0] = e0 * inv; alph[b * 3 + 1] = e1 * inv; alph[b * 3 + 2] = e2 * inv;
}

__global__ __launch_bounds__(256) void layernorm_concat_kernel(
    const float* __restrict__ P, const float* __restrict__ S, const float* __restrict__ Cm,
    const float* __restrict__ ln_g, const float* __restrict__ ln_b, float* __restrict__ Xn) {
  int b = blockIdx.y, i = blockIdx.x;
  long long rbase = ((long long)b * NN + i) * NN;
  float sum = 0.f, sq = 0.f;
  for (int t = threadIdx.x; t < IND; t += 256) {
    float v = (t < NN) ? P[rbase + t] : ((t < 2 * NN) ? S[rbase + t - NN] : Cm[rbase + t - 2 * NN]);
    sum += v; sq += v * v;
  }
  float ts = block_reduce_sum(sum);
  float tq = block_reduce_sum(sq);
  float mu = ts / (float)IND;
  float var = tq / (float)IND - mu * mu;
  float rstd = rsqrtf(var + 1e-5f);
  long long obase = ((long long)b * NN + i) * IND;
  for (int t = threadIdx.x; t < IND; t += 256) {
    float v = (t < NN) ? P[rbase + t] : ((t < 2 * NN) ? S[rbase + t - NN] : Cm[rbase + t - 2 * NN]);
    Xn[obase + t] = (v - mu) * rstd * ln_g[t] + ln_b[t];
  }
}

__global__ __launch_bounds__(256) void fuse_kernel(float* __restrict__ P,     // in/out (Afused)
                                                   const float* __restrict__ S,
                                                   const float* __restrict__ Cm,
                                                   const float* __restrict__ alph) {
  long long total = (long long)BB * NN * NN;
  for (long long idx = blockIdx.x * 256ll + threadIdx.x; idx < total;
       idx += (long long)gridDim.x * 256) {
    int j = (int)(idx % NN);
    int i = (int)((idx / NN) % NN);
    int b = (int)(idx / ((long long)NN * NN));
    float v = alph[b * 3 + 0] * P[idx] + alph[b * 3 + 1] * S[idx] + alph[b * 3 + 2] * Cm[idx];
    P[idx] = (i == j) ? 1.f : v;
  }
}

__global__ __launch_bounds__(256) void degree_kernel(const float* __restrict__ Af, float* dinv) {
  int b = blockIdx.y, i = blockIdx.x;
  const float* row = Af + ((long long)b * NN + i) * NN;
  float s = 0.f;
  for (int j = threadIdx.x; j < NN; j += 256) s += row[j];
  float r = block_reduce_sum(s);
  if (threadIdx.x == 0) dinv[b * NN + i] = rsqrtf(fmaxf(r, 1e-8f));
}

__global__ __launch_bounds__(256) void anorm_kernel(const float* __restrict__ Af,
                                                    const float* __restrict__ dinv,
                                                    float* __restrict__ An) {
  long long total = (long long)BB * NN * NN;
  for (long long idx = blockIdx.x * 256ll + threadIdx.x; idx < total;
       idx += (long long)gridDim.x * 256) {
    int j = (int)(idx % NN);
    int i = (int)((idx / NN) % NN);
    int b = (int)(idx / ((long long)NN * NN));
    An[idx] = dinv[b * NN + i] * Af[idx] * dinv[b * NN + j];
  }
}

__global__ __launch_bounds__(256) void colmean_kernel(const float* __restrict__ H, float* g) {
  int d = blockIdx.x * 256 + threadIdx.x, b = blockIdx.y;
  float a = 0.f;
  for (int i = 0; i < NN; ++i) a += H[((long long)b * NN + i) * NN + d];
  g[b * NN + d] = a * (1.f / (float)NN);
}

__global__ __launch_bounds__(256) void final_kernel(const float* __restrict__ g,
                                                    const float* __restrict__ Wc,
                                                    const float* __restrict__ bc,
                                                    float* __restrict__ out) {
  int b = blockIdx.x;
  float a = 0.f;
  for (int d = threadIdx.x; d < NN; d += 256) a += g[b * NN + d] * Wc[d];
  float r = block_reduce_sum(a);
  if (threadIdx.x == 0) out[b] = 1.f / (1.f + expf(-(r + bc[0])));
}

// ---------------- host orchestration ----------------
extern "C" void kernel_launch(void* const* d_in, const int* in_sizes, int n_in,
                              void* d_out, int out_size, void* d_ws, size_t ws_size,
                              hipStream_t stream) {
  const float* x    = (const float*)d_in[0];
  const float* gamma= (const float*)d_in[1];
  const float* ln_g = (const float*)d_in[2];
  const float* ln_b = (const float*)d_in[3];
  const float* W0   = (const float*)d_in[4];
  const float* b0   = (const float*)d_in[5];
  const float* W1   = (const float*)d_in[6];
  const float* b1   = (const float*)d_in[7];
  const float* W2   = (const float*)d_in[8];
  const float* b2   = (const float*)d_in[9];
  const float* Wc   = (const float*)d_in[10];
  const float* bc   = (const float*)d_in[11];
  float* out = (float*)d_out;

  float* ws = (float*)d_ws;
  const long long BTN = (long long)BB * TT * NN;       // 8,388,608
  const long long BNN = (long long)BB * NN * NN;       // 4,194,304
  // liveness-based layout (peak ~113 MiB)
  float* xm   = ws;                       // [0, BTN)
  float* rm   = ws + BTN;                 // [BTN, 2BTN)
  float* P    = ws + 2 * BTN;             // raw cov -> corr -> Afused (in place)
  float* S    = P + BNN;                  // raw cov -> corr ; later Anorm
  float* Cm   = S + BNN;                  // partial corr ; later H
  float* small= Cm + BNN;
  float* mu    = small;            // 8192
  float* sd    = mu + 8192;        // 8192
  float* rs    = sd + 8192;        // 8192
  float* scl   = rs + 8192;        // 16
  float* stats = scl + 16;         // 144
  float* alph  = stats + 144;      // 48
  float* dinv  = alph + 48;        // 8192
  float* gbuf  = dinv + 8192;      // 8192
  // reuse of xm/rm region once covariances done:
  float* Ainv = ws;                 // 4M floats
  float* Xk   = ws + BNN;           // 4M
  float* T1   = ws + 2 * BNN;       // 4M
  float* X1   = ws + 3 * BNN;       // 4M  (== 2*BTN boundary)
  // after inverse done:
  float* Xn = ws;                   // B*N*1536 = 12,582,912 floats
  float* Y  = ws + (long long)BB * NN * IND;  // 4M floats, ends at 2*BTN
  float* H  = Cm;                   // reuse C buffer
  float* Af = P;                    // fused adjacency in place of P
  float* An = S;                    // normalized adjacency in place of S

  const float covAlpha = 1.f / (1023.f + 1e-8f);
  dim3 gemmGrid(NN / BN, NN / BM, BB);
  const long long sNN = (long long)NN * NN;
  const long long sTN = (long long)TT * NN;

  // 1-3: center + ranks
  mean_kernel<<<dim3(NN / 64, BB), 256, 0, stream>>>(x, mu);
  center_kernel<<<4096, 256, 0, stream>>>(x, mu, xm);
  rank_kernel<<<dim3(NN, BB), 256, 0, stream>>>(x, rm);

  // 4-5: covariances via fp32 WMMA (A-transposed reads of xm / rm)
  gemm_wmma<true,false,false,false><<<gemmGrid, 256, 0, stream>>>(
      xm, NN, sTN, xm, NN, sTN, P, NN, sNN, nullptr, 0, nullptr, covAlpha, 0.f, TT);
  gemm_wmma<true,false,false,false><<<gemmGrid, 256, 0, stream>>>(
      rm, NN, sTN, rm, NN, sTN, S, NN, sNN, nullptr, 0, nullptr, covAlpha, 0.f, TT);

  // 6: ridge (raw P) -> Ainv ; then normalize P,S to correlations
  ridge_kernel<<<4096, 256, 0, stream>>>(P, Ainv);
  diag_sd_kernel<<<dim3(NN / 256, BB), 256, 0, stream>>>(P, sd);
  corr_norm_kernel<<<4096, 256, 0, stream>>>(P, sd);
  diag_sd_kernel<<<dim3(NN / 256, BB), 256, 0, stream>>>(S, sd);
  corr_norm_kernel<<<4096, 256, 0, stream>>>(S, sd);

  // 9-11: Newton-Schulz inverse (WMMA GEMMs)
  rowsum_abs_kernel<<<dim3(NN, BB), 256, 0, stream>>>(Ainv, rs);
  batch_scale_kernel<<<BB, 256, 0, stream>>>(rs, scl);
  scale_init_kernel<<<4096, 256, 0, stream>>>(Ainv, scl, Xk);
  float* Xa = Xk; float* Xb = X1;
  for (int it = 0; it < NS_ITERS; ++it) {
    gemm_wmma<false,false,false,false><<<gemmGrid, 256, 0, stream>>>(
        Ainv, NN, sNN, Xa, NN, sNN, T1, NN, sNN, nullptr, 0, nullptr, 1.f, 0.f, NN);
    gemm_wmma<false,true,false,false><<<gemmGrid, 256, 0, stream>>>(
        Xa, NN, sNN, T1, NN, sNN, Xb, NN, sNN, Xa, sNN, nullptr, -1.f, 2.f, NN);
    float* t = Xa; Xa = Xb; Xb = t;
  }
  // 12-13: partial correlation from precision matrix
  diag_sd_kernel<<<dim3(NN / 256, BB), 256, 0, stream>>>(Xa, sd);
  pcorr_kernel<<<4096, 256, 0, stream>>>(Xa, sd, Cm);

  // 14-16: view fusion weights
  zero_kernel<<<1, 256, 0, stream>>>(stats, 9 * BB);
  fusion_stats_kernel<<<dim3(64, BB), 256, 0, stream>>>(P, S, Cm, stats);
  alpha_kernel<<<1, 32, 0, stream>>>(stats, gamma, alph);

  // 17: LayerNorm(X=[P|S|C]) first (needs P), then fuse adjacency in place of P
  layernorm_concat_kernel<<<dim3(NN, BB), 256, 0, stream>>>(P, S, Cm, ln_g, ln_b, Xn);
  fuse_kernel<<<4096, 256, 0, stream>>>(Af, S, Cm, alph);

  // 18-19: adjacency normalization
  degree_kernel<<<dim3(NN, BB), 256, 0, stream>>>(Af, dinv);
  anorm_kernel<<<4096, 256, 0, stream>>>(Af, dinv, An);

  // 20: 3-layer GCN (WMMA GEMMs with bias / relu epilogues)
  gemm_wmma<false,false,true,false><<<gemmGrid, 256, 0, stream>>>(
      Xn, IND, (long long)NN * IND, W0, NN, 0, Y, NN, sNN, nullptr, 0, b0, 1.f, 0.f, IND);
  gemm_wmma<false,false,false,true><<<gemmGrid, 256, 0, stream>>>(
      An, NN, sNN, Y, NN, sNN, H, NN, sNN, nullptr, 0, nullptr, 1.f, 0.f, NN);
  gemm_wmma<false,false,true,false><<<gemmGrid, 256, 0, stream>>>(
      H, NN, sNN, W1, NN, 0, Y, NN, sNN, nullptr, 0, b1, 1.f, 0.f, NN);
  gemm_wmma<false,false,false,true><<<gemmGrid, 256, 0, stream>>>(
      An, NN, sNN, Y, NN, sNN, H, NN, sNN, nullptr, 0, nullptr, 1.f, 0.f, NN);
  gemm_wmma<false,false,true,false><<<gemmGrid, 256, 0, stream>>>(
      H, NN, sNN, W2, NN, 0, Y, NN, sNN, nullptr, 0, b2, 1.f, 0.f, NN);
  gemm_wmma<false,false,false,true><<<gemmGrid, 256, 0, stream>>>(
      An, NN, sNN, Y, NN, sNN, H, NN, sNN, nullptr, 0, nullptr, 1.f, 0.f, NN);

  // 21-22: readout
  colmean_kernel<<<dim3(NN / 256, BB), 256, 0, stream>>>(H, gbuf);
  final_kernel<<<BB, 256, 0, stream>>>(gbuf, Wc, bc, out);
  (void)in_sizes; (void)n_in; (void)out_size; (void)ws_size;
}